// PillarHist_58626303590419
// MI455X (gfx1250) — compile-verified
//
#include <hip/hip_runtime.h>

typedef __attribute__((ext_vector_type(2))) float v2f;
typedef __attribute__((ext_vector_type(8))) float v8f;

#define PH_NUM_BINS 64
#define PH_ZMIN     (-3.0f)
#define PH_INV_BIN  16.0f      // 1 / ((1-(-3))/64)
#define PH_K        192        // 3 * 64 bins
#define PH_LDH      196        // padded LDS row stride (floats), conflict-free A reads
#define PH_TILE_M   16

// ---------------------------------------------------------------------------
// Kernel A: per-16-pillar histogram (LDS atomics) + f32 WMMA GEMM + bias.
// Block = 128 threads = 4 waves; wave w computes output channels [16w, 16w+16).
// ---------------------------------------------------------------------------
__global__ __launch_bounds__(128) void pillar_hist_gemm(
    const float* __restrict__ feat,   // [M,64,4]
    const int*   __restrict__ npts,   // [M]
    const float* __restrict__ W,      // [64,192]
    const float* __restrict__ bias,   // [64]
    float*       __restrict__ xout,   // [M,64]
    int M)
{
  __shared__ float hist[PH_TILE_M * PH_LDH];
  const int tid = threadIdx.x;
  const int m0  = blockIdx.x * PH_TILE_M;

  // zero the histogram tile
  for (int idx = tid; idx < PH_TILE_M * PH_LDH; idx += 128) hist[idx] = 0.0f;
  __syncthreads();

  // phase 1: scatter-add (count, sum_z, sum_r) with LDS float atomics
  for (int p = tid; p < PH_TILE_M * 64; p += 128) {
    const int mi = p >> 6;          // pillar within tile
    const int n  = p & 63;          // point index
    const int m  = m0 + mi;
    if (m < M) {
      const float4 f = reinterpret_cast<const float4*>(feat)[(size_t)m * 64 + n];
      if (n < npts[m]) {
        const float z = f.z, r = f.w;
        int b = (int)((z - PH_ZMIN) * PH_INV_BIN);   // trunc-toward-zero like .astype(int32)
        b = b < 0 ? 0 : (b > 63 ? 63 : b);
        float* h = &hist[mi * PH_LDH + 3 * b];
        atomicAdd(h + 0, 1.0f);     // ds_add_f32
        atomicAdd(h + 1, z);
        atomicAdd(h + 2, r);
      }
    }
  }
  __syncthreads();

  // phase 2: sums -> means (interleaved [count, mean_z, mean_r] already in place)
  for (int q = tid; q < PH_TILE_M * PH_NUM_BINS; q += 128) {
    const int mi = q >> 6;
    const int b  = q & 63;
    float* h = &hist[mi * PH_LDH + 3 * b];
    const float inv = 1.0f / (h[0] + 1e-5f);
    h[1] *= inv;
    h[2] *= inv;
  }
  __syncthreads();

  // phase 3: x[16,64] = hist[16,192] @ W^T via V_WMMA_F32_16X16X4_F32
  // Fragment layout (ISA 7.12.2): lane = (h = lane>>4, i = lane&15)
  //   A 16x4 : lane holds A[i][2h + j] in vgpr j   (j = 0,1)
  //   B 4x16 : lane holds B[2h + j][i] in vgpr j
  //   C/D    : vgpr j holds D[j + 8h][i]
  const int wave = tid >> 5;        // N-tile (channels 16*wave ..)
  const int lane = tid & 31;
  const int hh   = lane >> 4;
  const int ii   = lane & 15;

  v8f acc = {};
  const float* wrow = W    + (size_t)(wave * 16 + ii) * PH_K + 2 * hh; // B[k][ii] = W[c][k]
  const float* arow = hist + ii * PH_LDH + 2 * hh;

#pragma unroll
  for (int k0 = 0; k0 < PH_K; k0 += 4) {
    const v2f a = *reinterpret_cast<const v2f*>(arow + k0);  // ds_load_b64
    const v2f b = *reinterpret_cast<const v2f*>(wrow + k0);  // global_load_b64 (L2-hot)
    acc = __builtin_amdgcn_wmma_f32_16x16x4_f32(
        /*neg_a=*/false, a, /*neg_b=*/false, b,
        /*c_mod=*/(short)0, acc, /*reuse_a=*/false, /*reuse_b=*/false);
  }

  const float bc = bias[wave * 16 + ii];
#pragma unroll
  for (int j = 0; j < 8; ++j) {
    const int m = m0 + j + 8 * hh;
    if (m < M) xout[(size_t)m * 64 + wave * 16 + ii] = acc[j] + bc;
  }
}

// ---------------------------------------------------------------------------
// Kernel B: per-block partial sums of x and x^2 per channel (deterministic).
// ---------------------------------------------------------------------------
__global__ __launch_bounds__(256) void bn_partial(
    const float* __restrict__ x, float* __restrict__ part, int M, int rowsPerBlk)
{
  __shared__ float ssum[256], ssq[256];
  const int t = threadIdx.x;
  const int c = t & 63;
  const int g = t >> 6;                  // 4 row groups
  const int rstart = blockIdx.x * rowsPerBlk;
  const int rend   = (rstart + rowsPerBlk < M) ? rstart + rowsPerBlk : M;
  float s = 0.0f, s2 = 0.0f;
  for (int r = rstart + g; r < rend; r += 4) {
    const float v = x[(size_t)r * 64 + c];
    s  += v;
    s2 += v * v;
  }
  ssum[t] = s; ssq[t] = s2;
  __syncthreads();
  if (t < 64) {
    const float a = ssum[t] + ssum[t + 64] + ssum[t + 128] + ssum[t + 192];
    const float b = ssq[t]  + ssq[t + 64]  + ssq[t + 128]  + ssq[t + 192];
    part[blockIdx.x * 128 + t]      = a;
    part[blockIdx.x * 128 + 64 + t] = b;
  }
}

// ---------------------------------------------------------------------------
// Kernel C: fold partials -> final 128 values (64 sums, 64 sums of squares).
// ---------------------------------------------------------------------------
__global__ __launch_bounds__(128) void bn_finish(
    const float* __restrict__ part, float* __restrict__ fin, int nblk)
{
  const int t = threadIdx.x;   // 0..127
  float s = 0.0f;
  for (int b = 0; b < nblk; ++b) s += part[b * 128 + t];
  fin[t] = s;
}

// ---------------------------------------------------------------------------
// Kernel D: in-place batchnorm (biased stats) + gamma/beta + ReLU.
// ---------------------------------------------------------------------------
__global__ __launch_bounds__(256) void bn_apply(
    float* __restrict__ x, const float* __restrict__ fin,
    const float* __restrict__ gamma, const float* __restrict__ beta, int M)
{
  const size_t idx = (size_t)blockIdx.x * 256 + threadIdx.x;
  if (idx >= (size_t)M * 64) return;
  const int   c    = (int)(idx & 63);
  const float invM = 1.0f / (float)M;
  const float mu   = fin[c] * invM;
  const float var  = fin[64 + c] * invM - mu * mu;
  const float istd = rsqrtf(var + 1e-5f);
  const float v = (x[idx] - mu) * istd * gamma[c] + beta[c];
  x[idx] = v > 0.0f ? v : 0.0f;
}

// ---------------------------------------------------------------------------
extern "C" void kernel_launch(void* const* d_in, const int* in_sizes, int n_in,
                              void* d_out, int out_size, void* d_ws, size_t ws_size,
                              hipStream_t stream) {
  const float* feat  = (const float*)d_in[0];  // features [M,64,4] f32
  const int*   npts  = (const int*)  d_in[1];  // num_points [M]
  // d_in[2] = coors (unused by reference math)
  const float* W     = (const float*)d_in[3];  // [64,192]
  const float* bias  = (const float*)d_in[4];  // [64]
  const float* gamma = (const float*)d_in[5];  // [64]
  const float* beta  = (const float*)d_in[6];  // [64]
  const int M = in_sizes[1];

  float* x    = (float*)d_out;                 // [M,64], normalized in place
  float* part = (float*)d_ws;                  // RB*128 floats
  const int RB = 250;
  const int rowsPerBlk = (M + RB - 1) / RB;
  float* fin  = part + RB * 128;               // 128 floats

  pillar_hist_gemm<<<(M + PH_TILE_M - 1) / PH_TILE_M, 128, 0, stream>>>(
      feat, npts, W, bias, x, M);
  bn_partial<<<RB, 256, 0, stream>>>(x, part, M, rowsPerBlk);
  bn_finish<<<1, 128, 0, stream>>>(part, fin, RB);
  bn_apply<<<(int)(((size_t)M * 64 + 255) / 256), 256, 0, stream>>>(
      x, fin, gamma, beta, M);
}